// pooling_layer_10599979286530
// MI455X (gfx1250) — compile-verified
//
#include <hip/hip_runtime.h>

// ---------------------------------------------------------------------------
// Fused (tokens @ W.T + bias) -> segment-max -> padded [B, M, D_OUT] scatter
// for MI455X (gfx1250, wave32, WMMA).
//
// Roofline: 52.4 GFLOP fp32, 819 MB token read (~35us HBM floor @ 23.3TB/s);
// f32 WMMA (K=4) makes the inner loop LDS-bandwidth bound. Each wave computes
// TWO 16-col N-tiles per A fragment (384B LDS / wmma). Block = 512 thr =
// 16 waves = 4 groups x 4 waves; each group owns one M-tile (4 in flight).
//
// NEW: software pipeline with DOUBLE-BUFFERED A sets. Per iteration:
//   issue global loads for tile t+1 -> compute tile t (128 wmma hide HBM)
//   -> ds_store prefetched data into the other set -> ONE barrier.
// Weight staged once per block in LDS; padded stride 260 keeps all WMMA
// fragment reads (ds_load_2addr_b64) and staging stores bank-conflict free.
// ---------------------------------------------------------------------------

typedef __attribute__((ext_vector_type(2))) float v2f;
typedef __attribute__((ext_vector_type(8))) float v8f;

#define D_IN        256
#define D_OUT       128
#define MAX_SENT    64                  // M in the reference
#define LDS_STRIDE  260                 // 256 + 4 pad: conflict-free frag loads
#define W_FLOATS    (D_OUT * LDS_STRIDE)            // 33280
#define A_OFF       W_FLOATS
#define ABUF_FLOATS (16 * LDS_STRIDE)               // 4160 per M-tile buffer
#define NGROUPS     4
#define SET_FLOATS  (NGROUPS * ABUF_FLOATS)         // 16640 per buffer set
#define LDS_FLOATS  (W_FLOATS + 2 * SET_FLOATS)     // 66560 fl = 266240 B

// --- kernel 0: zero the padded output -------------------------------------
__global__ void zero_out_kernel(float* __restrict__ out, int n) {
  int i = blockIdx.x * blockDim.x + threadIdx.x;
  if (i < n) out[i] = 0.0f;
}

// --- kernel 1: segment boundaries (seg_ids sorted, all segments >= 1) ------
__global__ void seg_bounds_kernel(const int* __restrict__ seg_ids,
                                  int* __restrict__ seg_start, int T, int S) {
  int t = blockIdx.x * blockDim.x + threadIdx.x;
  if (t >= T) return;
  int sid = seg_ids[t];
  if (t == 0 || seg_ids[t - 1] != sid) seg_start[sid] = t;
  if (t == T - 1) seg_start[S] = T;
}

// --- kernel 2: fused GEMM + segment max + scatter --------------------------
__global__ __launch_bounds__(512)
void sent_pool_kernel(const float* __restrict__ tokens,
                      const float* __restrict__ weight,
                      const float* __restrict__ bias,
                      const int* __restrict__ seg_start,
                      const int* __restrict__ sent_batch,
                      const int* __restrict__ sent_pos,
                      float* __restrict__ out) {
  extern __shared__ float lds[];

  const int tid   = threadIdx.x;
  const int lane  = tid & 31;
  const int wave  = tid >> 5;        // 0..15
  const int grp   = wave >> 2;       // 0..3 : which M-tile of the 4 in flight
  const int wig   = wave & 3;        // 0..3 : wave inside group
  const int n0    = wig * 32;        // wave owns output cols [n0,n0+32)
  const int s     = blockIdx.x;      // sentence id

  // ---- stage full weight matrix (row-major, padded stride) into LDS ------
#pragma unroll 4
  for (int it = 0; it < (D_OUT * D_IN) / (512 * 4); ++it) {
    int idx = (it * 512 + tid) * 4;
    int row = idx >> 8;
    int col = idx & (D_IN - 1);
    const float4 w = *(const float4*)(weight + idx);
    *(float4*)(&lds[row * LDS_STRIDE + col]) = w;
  }

  const int s_beg = seg_start[s];
  const int s_end = seg_start[s + 1];
  const int len   = s_end - s_beg;
  const int nTiles = (len + 63) >> 6;   // outer iterations (64 rows each)

  const int hi = lane >> 4;          // lane half (rows +8, cols k+2)
  const int lm = lane & 15;

  // ---- prologue: stage tile 0 into buffer set 0 ---------------------------
#pragma unroll
  for (int it = 0; it < 8; ++it) {
    int idx = (it * 512 + tid) * 4;
    int row = idx >> 8;              // 0..63
    int col = idx & (D_IN - 1);
    int buf = row >> 4;
    int r16 = row & 15;
    float4 v = make_float4(0.f, 0.f, 0.f, 0.f);
    if (row < len)
      v = *(const float4*)(tokens + (size_t)(s_beg + row) * D_IN + col);
    *(float4*)(&lds[A_OFF + buf * ABUF_FLOATS + r16 * LDS_STRIDE + col]) = v;
  }
  __syncthreads();

  float cmax0 = -__builtin_inff();   // running max, cols n0 + lm
  float cmax1 = -__builtin_inff();   // running max, cols n0 + 16 + lm

  const float* aBase0 = &lds[A_OFF + grp * ABUF_FLOATS + lm * LDS_STRIDE + 2 * hi];
  const float* aBase1 = aBase0 + SET_FLOATS;
  const float* b0Base = &lds[(n0 + lm) * LDS_STRIDE + 2 * hi];
  const float* b1Base = &lds[(n0 + 16 + lm) * LDS_STRIDE + 2 * hi];

  for (int t = 0; t < nTiles; ++t) {
    const int cur = t & 1;
    const int m0  = t << 6;
    const bool havePf = (t + 1 < nTiles);

    // -- issue global loads for tile t+1 (latency hidden by the wmma loop) --
    float4 pf[8];
    if (havePf) {
      const int pm0 = m0 + 64;
#pragma unroll
      for (int it = 0; it < 8; ++it) {
        int idx = (it * 512 + tid) * 4;
        int row = idx >> 8;
        int col = idx & (D_IN - 1);
        pf[it] = make_float4(0.f, 0.f, 0.f, 0.f);
        if (pm0 + row < len)
          pf[it] = *(const float4*)(tokens + (size_t)(s_beg + pm0 + row) * D_IN + col);
      }
    }

    // -- compute tile t from set `cur` --------------------------------------
    const int tileBase = m0 + grp * 16;
    if (tileBase < len) {            // wave-uniform: skip dead tail tiles
      const float* aB = cur ? aBase1 : aBase0;
      v8f acc0 = {};
      v8f acc1 = {};
#pragma unroll
      for (int k = 0; k < D_IN; k += 4) {
        v2f a  = *(const v2f*)(aB + k);
        v2f b0 = *(const v2f*)(b0Base + k);
        v2f b1 = *(const v2f*)(b1Base + k);
        acc0 = __builtin_amdgcn_wmma_f32_16x16x4_f32(
            false, a, false, b0, (short)0, acc0, false, false);
        acc1 = __builtin_amdgcn_wmma_f32_16x16x4_f32(
            false, a, false, b1, (short)0, acc1, false, false);
      }
      // Fold valid rows (D layout: reg v = row v, +8 for upper lane half).
#pragma unroll
      for (int v = 0; v < 8; ++v) {
        if (tileBase + v + 8 * hi < len) {
          cmax0 = fmaxf(cmax0, acc0[v]);
          cmax1 = fmaxf(cmax1, acc1[v]);
        }
      }
    }

    // -- store prefetched tile t+1 into the other set (disjoint region) -----
    if (havePf) {
      const int so = A_OFF + (cur ^ 1) * SET_FLOATS;
#pragma unroll
      for (int it = 0; it < 8; ++it) {
        int idx = (it * 512 + tid) * 4;
        int row = idx >> 8;
        int col = idx & (D_IN - 1);
        int buf = row >> 4;
        int r16 = row & 15;
        *(float4*)(&lds[so + buf * ABUF_FLOATS + r16 * LDS_STRIDE + col]) = pf[it];
      }
    }
    __syncthreads();                 // single barrier per iteration
  }

  // Combine row halves: lane L holds rows {0..7}, lane L+16 rows {8..15}.
  cmax0 = fmaxf(cmax0, __shfl_xor(cmax0, 16, 32));
  cmax1 = fmaxf(cmax1, __shfl_xor(cmax1, 16, 32));

  // Cross-group reduction through LDS (alias the dead A region).
  float* red = &lds[A_OFF];          // 4 groups x 128 cols
  if (lane < 16) {
    red[grp * D_OUT + n0 + lane]      = cmax0;
    red[grp * D_OUT + n0 + 16 + lane] = cmax1;
  }
  __syncthreads();

  if (tid < D_OUT) {
    float m = red[tid];
#pragma unroll
    for (int g = 1; g < NGROUPS; ++g) m = fmaxf(m, red[g * D_OUT + tid]);
    const int ob = sent_batch[s];
    const int op = sent_pos[s];
    out[((size_t)ob * MAX_SENT + op) * D_OUT + tid] = m + bias[tid];
  }
}

// ---------------------------------------------------------------------------
extern "C" void kernel_launch(void* const* d_in, const int* in_sizes, int n_in,
                              void* d_out, int out_size, void* d_ws, size_t ws_size,
                              hipStream_t stream) {
  const float* tokens     = (const float*)d_in[0];
  const float* weight     = (const float*)d_in[1];
  const float* bias       = (const float*)d_in[2];
  const int*   seg_ids    = (const int*)d_in[3];
  const int*   sent_batch = (const int*)d_in[4];
  const int*   sent_pos   = (const int*)d_in[5];
  float*       out        = (float*)d_out;

  const int T = in_sizes[3];
  const int S = in_sizes[4];

  int* seg_start = (int*)d_ws;   // (S+1) ints of scratch

  zero_out_kernel<<<(out_size + 255) / 256, 256, 0, stream>>>(out, out_size);
  seg_bounds_kernel<<<(T + 255) / 256, 256, 0, stream>>>(seg_ids, seg_start, T, S);

  const size_t ldsBytes = (size_t)LDS_FLOATS * sizeof(float);  // 266240 B
  sent_pool_kernel<<<S, 512, ldsBytes, stream>>>(
      tokens, weight, bias, seg_start, sent_batch, sent_pos, out);
}